// SoftmaxProbs_89172111000255
// MI455X (gfx1250) — compile-verified
//
#include <hip/hip_runtime.h>
#include <stdint.h>

#define NCLS 18
#define BLOCK 256
#define ROWS_PER_TILE 256
#define TILE_FLOATS (ROWS_PER_TILE * NCLS) /* 4608 floats = 18 KB */
#define NF4_FULL (TILE_FLOATS / 4)         /* 1152 */
#define STAGE_ITERS 5                      /* ceil(1152 / 256) */
#define ASYNC_PER_TILE (2 * STAGE_ITERS)   /* wave-uniform async ops per tile */
#define NBLOCKS 1024

#if defined(__AMDGCN__) && __has_builtin(__builtin_amdgcn_global_load_async_to_lds_b128)
#define USE_ASYNC 1
#else
#define USE_ASYNC 0
#endif

#if USE_ASYNC
typedef int v4i __attribute__((vector_size(4 * sizeof(int))));
typedef __attribute__((address_space(1))) v4i* gvec_ptr;
typedef __attribute__((address_space(3))) v4i* lvec_ptr;
#endif

__device__ __forceinline__ void wait_async_le(int) {}  // placeholder overload guard

__device__ __forceinline__ void wait_async_prev_tile_done() {
#if USE_ASYNC
#if __has_builtin(__builtin_amdgcn_s_wait_asynccnt)
  __builtin_amdgcn_s_wait_asynccnt(ASYNC_PER_TILE);
#else
  asm volatile("s_wait_asynccnt %0" ::"i"(ASYNC_PER_TILE) : "memory");
#endif
#endif
}

__device__ __forceinline__ void wait_async_all_done() {
#if USE_ASYNC
#if __has_builtin(__builtin_amdgcn_s_wait_asynccnt)
  __builtin_amdgcn_s_wait_asynccnt(0);
#else
  asm volatile("s_wait_asynccnt 0" ::: "memory");
#endif
#endif
}

// Stage one tile (both arrays) into LDS with coalesced 128-bit async
// global->LDS copies. Fixed 5 iterations x 2 loads so every wave issues
// exactly ASYNC_PER_TILE async ops (chunk index clamped for the remainder:
// duplicate DMA writes of identical bytes, benign + deterministic).
__device__ __forceinline__ void stage_tile(const float* __restrict__ gx,
                                           const float* __restrict__ gl,
                                           float* __restrict__ bx,
                                           float* __restrict__ bl, int nf4,
                                           int nflt, int tid) {
#if USE_ASYNC
#pragma unroll
  for (int k = 0; k < STAGE_ITERS; ++k) {
    const int i = tid + k * BLOCK;
    const int j = (i < nf4) ? i : (nf4 - 1);
    __builtin_amdgcn_global_load_async_to_lds_b128(
        (gvec_ptr)(v4i*)(void*)(gx + 4 * j),
        (lvec_ptr)(v4i*)(void*)(bx + 4 * j), 0, 0);
    __builtin_amdgcn_global_load_async_to_lds_b128(
        (gvec_ptr)(v4i*)(void*)(gl + 4 * j),
        (lvec_ptr)(v4i*)(void*)(bl + 4 * j), 0, 0);
  }
#else
  for (int i = tid; i < nf4; i += BLOCK) {
    float4 vx = *(const float4*)(gx + 4 * i);
    float4 vl = *(const float4*)(gl + 4 * i);
    *(float4*)(bx + 4 * i) = vx;
    *(float4*)(bl + 4 * i) = vl;
  }
#endif
  // scalar tail (only if validRows*NCLS not divisible by 4)
  for (int i = (nf4 << 2) + tid; i < nflt; i += BLOCK) {
    bx[i] = gx[i];
    bl[i] = gl[i];
  }
}

__device__ __forceinline__ float row_loss(const float* __restrict__ bx,
                                          const float* __restrict__ bl) {
  float xv[NCLS];
#pragma unroll
  for (int c = 0; c < NCLS; ++c) xv[c] = bx[c];
  float m = xv[0];
#pragma unroll
  for (int c = 1; c < NCLS; ++c) m = fmaxf(m, xv[c]);
  float s = 0.0f;
#pragma unroll
  for (int c = 0; c < NCLS; ++c) s += __expf(xv[c] - m);
  float sl = 0.0f, sx = 0.0f;
#pragma unroll
  for (int c = 0; c < NCLS; ++c) {
    const float l = bl[c];
    sl += l;
    sx = fmaf(l, xv[c], sx);
  }
  // sum_c l*(logZ - x) = logZ*sum(l) - sum(l*x)
  return (m + __logf(s)) * sl - sx;
}

__global__ __launch_bounds__(BLOCK) void ce_partial_kernel(
    const float* __restrict__ logits, const float* __restrict__ labels,
    float* __restrict__ partial, int B) {
  __shared__ float lds_x[2][TILE_FLOATS];
  __shared__ float lds_l[2][TILE_FLOATS];
  __shared__ float red[BLOCK / 32];

  const int tid = (int)threadIdx.x;
  const int numTiles = (B + ROWS_PER_TILE - 1) / ROWS_PER_TILE;

  float acc = 0.0f;
  int t = (int)blockIdx.x;
  int sel = 0;

  // prologue: prefetch first tile into buffer 0
  if (t < numTiles) {
    const long long base = (long long)t * TILE_FLOATS;
    const int vr = min(ROWS_PER_TILE, B - t * ROWS_PER_TILE);
    stage_tile(logits + base, labels + base, lds_x[0], lds_l[0],
               (vr * NCLS) >> 2, vr * NCLS, tid);
  }

  for (; t < numTiles; t += (int)gridDim.x) {
    const int tn = t + (int)gridDim.x;
    const bool pre = (tn < numTiles);  // uniform across the block

    // prefetch next tile into the other buffer while current is in flight
    if (pre) {
      const long long nb = (long long)tn * TILE_FLOATS;
      const int vrn = min(ROWS_PER_TILE, B - tn * ROWS_PER_TILE);
      stage_tile(logits + nb, labels + nb, lds_x[sel ^ 1], lds_l[sel ^ 1],
                 (vrn * NCLS) >> 2, vrn * NCLS, tid);
      // drain exactly the current tile's copies; next tile's stay in flight
      wait_async_prev_tile_done();
    } else {
      wait_async_all_done();
    }
    __syncthreads();

    const int vr = min(ROWS_PER_TILE, B - t * ROWS_PER_TILE);
    if (tid < vr)
      acc += row_loss(&lds_x[sel][tid * NCLS], &lds_l[sel][tid * NCLS]);
    __syncthreads();  // everyone done reading buf[sel] before it is reused
    sel ^= 1;
  }

  // ---- deterministic block reduction (wave32 shuffle tree + LDS) ----
#pragma unroll
  for (int off = 16; off > 0; off >>= 1) acc += __shfl_down(acc, off, 32);
  const int lane = tid & 31;
  const int wid = tid >> 5;
  if (lane == 0) red[wid] = acc;
  __syncthreads();
  if (tid == 0) {
    float s = 0.0f;
#pragma unroll
    for (int w = 0; w < BLOCK / 32; ++w) s += red[w];
    partial[blockIdx.x] = s;
  }
}

__global__ __launch_bounds__(256) void ce_final_kernel(
    const float* __restrict__ partial, float* __restrict__ out, int nb,
    float invB) {
  __shared__ float red[8];
  float s = 0.0f;
  for (int i = (int)threadIdx.x; i < nb; i += 256) s += partial[i];
#pragma unroll
  for (int off = 16; off > 0; off >>= 1) s += __shfl_down(s, off, 32);
  const int lane = (int)threadIdx.x & 31;
  const int wid = (int)threadIdx.x >> 5;
  if (lane == 0) red[wid] = s;
  __syncthreads();
  if (threadIdx.x == 0) {
    float t = 0.0f;
#pragma unroll
    for (int w = 0; w < 8; ++w) t += red[w];
    out[0] = t * invB;
  }
}

extern "C" void kernel_launch(void* const* d_in, const int* in_sizes, int n_in,
                              void* d_out, int out_size, void* d_ws,
                              size_t ws_size, hipStream_t stream) {
  const float* logits = (const float*)d_in[0];
  const float* labels = (const float*)d_in[1];
  float* out = (float*)d_out;
  float* partial = (float*)d_ws;  // NBLOCKS floats = 4 KB scratch

  const int B = in_sizes[0] / NCLS;

  ce_partial_kernel<<<NBLOCKS, BLOCK, 0, stream>>>(logits, labels, partial, B);
  ce_final_kernel<<<1, 256, 0, stream>>>(partial, out, NBLOCKS,
                                         (float)(1.0 / (double)B));
}